// SparseCoder_14740327760019
// MI455X (gfx1250) — compile-verified
//
#include <hip/hip_runtime.h>

#define BATCH 256
#define EPB   256   // edges staged per block

typedef __attribute__((ext_vector_type(2))) float v2f;
typedef __attribute__((ext_vector_type(8))) float v8f;

// ---------------- transpose (BATCH x n) -> (n x BATCH) ----------------
__global__ void k_transpose_in(const float* __restrict__ in, float* __restrict__ out, int n) {
    __shared__ float tile[32][33];
    const int bx = blockIdx.x * 32;   // feature
    const int by = blockIdx.y * 32;   // batch
    const int tx = threadIdx.x, ty = threadIdx.y;
    for (int i = ty; i < 32; i += 8) {
        int c = bx + tx;
        tile[i][tx] = (c < n) ? in[(by + i) * (long)n + c] : 0.f;
    }
    __syncthreads();
    for (int i = ty; i < 32; i += 8) {
        int c = bx + i;
        if (c < n) out[(long)c * BATCH + (by + tx)] = tile[tx][i];
    }
}

// ---------------- transpose (n x BATCH) -> (BATCH x n) ----------------
__global__ void k_transpose_out(const float* __restrict__ in, float* __restrict__ out, int n) {
    __shared__ float tile[32][33];
    const int bx = blockIdx.x * 32;   // feature
    const int by = blockIdx.y * 32;   // batch
    const int tx = threadIdx.x, ty = threadIdx.y;
    for (int i = ty; i < 32; i += 8) {
        int c = bx + i;
        if (c < n) tile[i][tx] = in[(long)c * BATCH + (by + tx)];
    }
    __syncthreads();
    for (int i = ty; i < 32; i += 8) {
        int c = bx + tx;
        if (c < n) out[(by + i) * (long)n + c] = tile[tx][i];
    }
}

// ---------------- bias broadcast via WMMA rank-1 outer product ----------------
// yT[(r0+m)*BATCH + b0+n] = bias[r0+m] for a 16x16 tile, one wave32 per tile.
// A[m][k] = bias[m] (replicated over all K), B[k][n] = 0.25  =>  D[m][n] = bias[m].
__global__ __launch_bounds__(32) void k_bias_init_wmma(const float* __restrict__ bias,
                                                       float* __restrict__ yT, int n_out) {
    const int lane = threadIdx.x;           // 0..31
    const int r0 = blockIdx.x * 16;
    const int b0 = blockIdx.y * 16;
    const int m16 = lane & 15;
    const int ridx = r0 + m16;
    const float bv = (ridx < n_out) ? bias[ridx] : 0.f;

    v2f a = {bv, bv};                       // lane L holds A[M=L&15][*] for its two K slots
    v2f b = {0.25f, 0.25f};                 // all-0.25 B
    v8f c = {};
    v8f d = __builtin_amdgcn_wmma_f32_16x16x4_f32(
        /*neg_a=*/false, a, /*neg_b=*/false, b,
        /*c_mod=*/(short)0, c, /*reuse_a=*/false, /*reuse_b=*/false);

    const int mhi = (lane >> 4) << 3;       // D: vgpr v, lanes>=16 -> M = v + 8
#pragma unroll
    for (int v = 0; v < 8; ++v) {
        int r = r0 + v + mhi;
        if (r < n_out) yT[(long)r * BATCH + b0 + m16] = d[v];
    }
}

// ---------------- edge-parallel scatter: yT[r,:] += vals[e] * relu?(xT[c,:]) ----------------
__global__ __launch_bounds__(256) void k_scatter(const int* __restrict__ rows,
                                                 const int* __restrict__ cols,
                                                 const float* __restrict__ vals,
                                                 const float* __restrict__ xT,
                                                 float* __restrict__ yT,
                                                 int E, int relu_in) {
    __shared__ __align__(16) int   s_rows[EPB];
    __shared__ __align__(16) int   s_cols[EPB];
    __shared__ __align__(16) float s_vals[EPB];

    const int tid = threadIdx.x;            // owns batch column `tid`
    const int e0  = blockIdx.x * EPB;
    const int cnt = min(EPB, E - e0);       // E%EPB is 0 or 128; cnt % 4 == 0 always

    // Async staging of the edge chunk into LDS (CDNA5 async-to-LDS path, ASYNCcnt).
    const int nb = (cnt * 4) >> 4;          // 16B packets per array (<= 64)
    if (tid < nb) {
        unsigned lr = (unsigned)(uintptr_t)(&s_rows[0]) + (unsigned)tid * 16u;
        unsigned lc = (unsigned)(uintptr_t)(&s_cols[0]) + (unsigned)tid * 16u;
        unsigned lv = (unsigned)(uintptr_t)(&s_vals[0]) + (unsigned)tid * 16u;
        unsigned voff = (unsigned)tid * 16u;
        unsigned long long gr = (unsigned long long)(uintptr_t)(rows + e0);
        unsigned long long gc = (unsigned long long)(uintptr_t)(cols + e0);
        unsigned long long gv = (unsigned long long)(uintptr_t)(vals + e0);
        asm volatile("global_load_async_to_lds_b128 %0, %1, %2" :: "v"(lr), "v"(voff), "s"(gr) : "memory");
        asm volatile("global_load_async_to_lds_b128 %0, %1, %2" :: "v"(lc), "v"(voff), "s"(gc) : "memory");
        asm volatile("global_load_async_to_lds_b128 %0, %1, %2" :: "v"(lv), "v"(voff), "s"(gv) : "memory");
    }
    asm volatile("s_wait_asynccnt 0x0" ::: "memory");
    __syncthreads();

#pragma unroll 4
    for (int e = 0; e < cnt; ++e) {
        const int   c = s_cols[e];          // LDS broadcast reads
        const int   r = s_rows[e];
        const float w = s_vals[e];
        float xv = xT[(long)c * BATCH + tid];           // coalesced 1KB row, L2-resident
        if (relu_in) xv = fmaxf(xv, 0.f);
        __hip_atomic_fetch_add(&yT[(long)r * BATCH + tid], w * xv,
                               __ATOMIC_RELAXED, __HIP_MEMORY_SCOPE_AGENT);
    }
}

// ---------------- launch ----------------
extern "C" void kernel_launch(void* const* d_in, const int* in_sizes, int n_in,
                              void* d_out, int out_size, void* d_ws, size_t ws_size,
                              hipStream_t stream) {
    const float* x     = (const float*)d_in[0];
    const int*   rows0 = (const int*)  d_in[1];
    const int*   cols0 = (const int*)  d_in[2];
    const float* vals0 = (const float*)d_in[3];
    const float* bias0 = (const float*)d_in[4];
    const int*   rows1 = (const int*)  d_in[5];
    const int*   cols1 = (const int*)  d_in[6];
    const float* vals1 = (const float*)d_in[7];
    const float* bias1 = (const float*)d_in[8];
    const int*   rows2 = (const int*)  d_in[9];
    const int*   cols2 = (const int*)  d_in[10];
    const float* vals2 = (const float*)d_in[11];
    const float* bias2 = (const float*)d_in[12];

    char* ws = (char*)d_ws;
    float* xT  = (float*)(ws);                                     // 20000*256*4 = 20.48 MB
    float* y1T = (float*)(ws + 20480000);                          //  8000*256*4 =  8.19 MB
    float* y2T = (float*)(ws + 20480000 + 8192000);                //  2000*256*4 =  2.05 MB
    float* y3T = (float*)(ws + 20480000 + 8192000 + 2048000);      //   500*256*4 =  0.51 MB

    dim3 tb(32, 8);

    // input (256 x 20000) -> feature-major (20000 x 256)
    k_transpose_in<<<dim3((20000 + 31) / 32, BATCH / 32), tb, 0, stream>>>(x, xT, 20000);

    // layer 0: 20000 -> 8000
    k_bias_init_wmma<<<dim3((8000 + 15) / 16, BATCH / 16), 32, 0, stream>>>(bias0, y1T, 8000);
    k_scatter<<<(256000 + EPB - 1) / EPB, 256, 0, stream>>>(rows0, cols0, vals0, xT, y1T, 256000, 0);

    // layer 1: 8000 -> 2000 (ReLU folded into reads of y1T)
    k_bias_init_wmma<<<dim3((2000 + 15) / 16, BATCH / 16), 32, 0, stream>>>(bias1, y2T, 2000);
    k_scatter<<<(64000 + EPB - 1) / EPB, 256, 0, stream>>>(rows1, cols1, vals1, y1T, y2T, 64000, 1);

    // layer 2: 2000 -> 500
    k_bias_init_wmma<<<dim3((500 + 15) / 16, BATCH / 16), 32, 0, stream>>>(bias2, y3T, 500);
    k_scatter<<<(16000 + EPB - 1) / EPB, 256, 0, stream>>>(rows2, cols2, vals2, y2T, y3T, 16000, 1);

    // (500 x 256) -> output (256 x 500)
    k_transpose_out<<<dim3((500 + 31) / 32, BATCH / 32), tb, 0, stream>>>(y3T, (float*)d_out, 500);
}